// Attn_6545530159401
// MI455X (gfx1250) — compile-verified
//
#include <hip/hip_runtime.h>
#include <math.h>

typedef __attribute__((ext_vector_type(2))) float v2f;
typedef __attribute__((ext_vector_type(8))) float v8f;

#define T_DIM 2048
#define B_DIM 32
#define H_DIM 512
#define M_DIM (T_DIM * B_DIM)   // 65536 GEMM rows, row r = t*B + b (memory order)
#define KC 32                   // K chunk staged in LDS
#define LDA 36                  // padded LDS strides: float4-aligned, bank-conflict-free
#define LDB 36

// ---------------------------------------------------------------------------
// Kernel 1: pre[b][o] = b_attn[o] + sum_h hidden[b][h] * W_attn[o][h]   (W1 part)
// ---------------------------------------------------------------------------
__global__ void precompute_kernel(const float* __restrict__ hidden,
                                  const float* __restrict__ W,     // (512, 1024)
                                  const float* __restrict__ bias,  // (512)
                                  float* __restrict__ pre)         // (32, 512)
{
    int idx = blockIdx.x * blockDim.x + threadIdx.x;  // 0..16383
    int b = idx >> 9;
    int o = idx & 511;
    const float4* hp = (const float4*)(hidden + b * H_DIM);
    const float4* wp = (const float4*)(W + (size_t)o * (2 * H_DIM));
    float s = 0.f;
    for (int i = 0; i < H_DIM / 4; ++i) {
        float4 hv = hp[i];
        float4 wv = wp[i];
        s += hv.x * wv.x + hv.y * wv.y + hv.z * wv.z + hv.w * wv.w;
    }
    pre[idx] = s + bias[o];
}

// ---------------------------------------------------------------------------
// Kernel 2: fused  scores[b][t] = sum_o tanh( enc_row . W2[o] + pre[b][o] ) * v[o]
// via fp32 WMMA (v_wmma_f32_16x16x4_f32). Workgroup: 64 rows x all 512 cols.
// 16 waves arranged 4(M) x 4(N); wave tile = 16 x 128 (8 accum tiles).
// Global->register prefetch of chunk i+1 overlaps the WMMA loop of chunk i.
// ---------------------------------------------------------------------------
__launch_bounds__(512, 1)
__global__ void attn_energy_kernel(const float* __restrict__ enc,   // (65536, 512)
                                   const float* __restrict__ W,     // (512, 1024); W2 = cols [512..1024)
                                   const float* __restrict__ pre,   // (32, 512)
                                   const float* __restrict__ v,     // (512)
                                   float* __restrict__ scores)      // (32, 2048)
{
    __shared__ float ldsA[64 * LDA];    //  9 KB : A tile, 64 rows x KC
    __shared__ float ldsB[512 * LDB];   // 72 KB : B tile (K-transposed), 512 n x KC
    __shared__ float ldsR[4][64];       //  1 KB : cross-wave score reduction

    const int tid  = threadIdx.x;
    const int lane = tid & 31;
    const int wave = tid >> 5;          // 0..15
    const int wm   = wave & 3;          // M sub-tile (16 rows each)
    const int wn   = wave >> 2;         // N sub-range (128 cols each)
    const int lo   = lane & 15;
    const int hi   = lane >> 4;         // half-wave select

    const int rowBase = blockIdx.x * 64;

    // cooperative-load coordinates (fixed per thread)
    const int ar  = tid >> 3;                 // A row 0..63
    const int ac4 = (tid & 7) * 4;            // A col group
    const float* aSrcBase = enc + (size_t)(rowBase + ar) * H_DIM + ac4;

    v8f acc[8];
#pragma unroll
    for (int j = 0; j < 8; ++j) acc[j] = (v8f)(0.f);

    // prefetch registers: 1 A float4 + 8 B float4 per thread
    float4 pa;
    float4 pb[8];

    // --- issue global loads for chunk at k0 into registers ---
    auto issue_loads = [&](int k0) {
        pa = *(const float4*)(aSrcBase + k0);
#pragma unroll
        for (int i = 0; i < 8; ++i) {
            int linear = tid + i * 512;
            int n   = linear >> 3;
            int c4  = (linear & 7) * 4;
            pb[i] = *(const float4*)(W + (size_t)n * 1024 + 512 + k0 + c4);
        }
    };
    // --- commit prefetched registers to LDS ---
    auto store_lds = [&]() {
        *(float4*)(&ldsA[ar * LDA + ac4]) = pa;
#pragma unroll
        for (int i = 0; i < 8; ++i) {
            int linear = tid + i * 512;
            int n  = linear >> 3;
            int c4 = (linear & 7) * 4;
            *(float4*)(&ldsB[n * LDB + c4]) = pb[i];
        }
    };

    issue_loads(0);
    store_lds();
    __syncthreads();

    for (int k0 = 0; k0 < H_DIM; k0 += KC) {
        const bool hasNext = (k0 + KC) < H_DIM;
        if (hasNext) issue_loads(k0 + KC);   // in flight during the WMMA loop below

        // 8 K-steps of 4; per step: 1 A frag + 8 B frags + 8 WMMAs
#pragma unroll
        for (int kk = 0; kk < KC; kk += 4) {
            int col = kk + 2 * hi;   // A 16x4 f32 layout: lane holds K = 2*hi + {0,1}
            v2f afrag = *(const v2f*)(&ldsA[(wm * 16 + lo) * LDA + col]);
#pragma unroll
            for (int j = 0; j < 8; ++j) {
                int n = wn * 128 + j * 16 + lo;
                v2f bfrag = *(const v2f*)(&ldsB[n * LDB + col]);
                acc[j] = __builtin_amdgcn_wmma_f32_16x16x4_f32(
                    false, afrag, false, bfrag, (short)0, acc[j], false, false);
            }
        }

        __syncthreads();             // all waves done reading current LDS tile
        if (hasNext) {
            store_lds();             // waits loadcnt here, not before compute
            __syncthreads();
        }
    }

    // Epilogue: energy = tanh(acc + pre[b][n]); partial score = energy * v[n]
    // C/D layout: VGPR i, lanes 0-15 -> M=i, lanes 16-31 -> M=i+8; N = lane&15.
    float part[8];
#pragma unroll
    for (int i = 0; i < 8; ++i) part[i] = 0.f;
#pragma unroll
    for (int j = 0; j < 8; ++j) {
        int n = wn * 128 + j * 16 + lo;
        float vn = v[n];
#pragma unroll
        for (int i = 0; i < 8; ++i) {
            // rowBase is a multiple of 64 -> batch index is block-independent
            int b = (wm * 16 + i + 8 * hi) & 31;
            float e = tanhf(acc[j][i] + pre[b * H_DIM + n]);
            part[i] += e * vn;
        }
    }
    // Reduce over the 16 lanes of each half (xor 1,2,4,8 stays within the half)
#pragma unroll
    for (int i = 0; i < 8; ++i) {
        float s = part[i];
        s += __shfl_xor(s, 1, 32);
        s += __shfl_xor(s, 2, 32);
        s += __shfl_xor(s, 4, 32);
        s += __shfl_xor(s, 8, 32);
        part[i] = s;
    }
    __syncthreads();
    if (lo == 0) {
#pragma unroll
        for (int i = 0; i < 8; ++i) {
            int mlocal = wm * 16 + i + 8 * hi;
            ldsR[wn][mlocal] = part[i];
        }
    }
    __syncthreads();
    if (tid < 64) {
        float s = ldsR[0][tid] + ldsR[1][tid] + ldsR[2][tid] + ldsR[3][tid];
        int gm = rowBase + tid;
        int b = gm & 31;
        int t = gm >> 5;
        scores[b * T_DIM + t] = s;
    }
}

// ---------------------------------------------------------------------------
// Kernel 3: softmax over T per batch; out[b][0][t]
// ---------------------------------------------------------------------------
__global__ void softmax_kernel(const float* __restrict__ scores,
                               float* __restrict__ out)
{
    __shared__ float red[256];
    int b = blockIdx.x;
    int tid = threadIdx.x;
    const float* s = scores + b * T_DIM;

    float mx = -INFINITY;
    for (int t = tid; t < T_DIM; t += 256) mx = fmaxf(mx, s[t]);
    red[tid] = mx;
    __syncthreads();
    for (int o = 128; o > 0; o >>= 1) {
        if (tid < o) red[tid] = fmaxf(red[tid], red[tid + o]);
        __syncthreads();
    }
    mx = red[0];
    __syncthreads();

    float ex[8];
    float sum = 0.f;
#pragma unroll
    for (int i = 0; i < 8; ++i) {
        ex[i] = expf(s[tid + i * 256] - mx);
        sum += ex[i];
    }
    red[tid] = sum;
    __syncthreads();
    for (int o = 128; o > 0; o >>= 1) {
        if (tid < o) red[tid] += red[tid + o];
        __syncthreads();
    }
    float inv = 1.f / red[0];
#pragma unroll
    for (int i = 0; i < 8; ++i)
        out[b * T_DIM + tid + i * 256] = ex[i] * inv;
}

// ---------------------------------------------------------------------------
extern "C" void kernel_launch(void* const* d_in, const int* in_sizes, int n_in,
                              void* d_out, int out_size, void* d_ws, size_t ws_size,
                              hipStream_t stream)
{
    const float* hidden = (const float*)d_in[0];  // (1,32,512)
    const float* enc    = (const float*)d_in[1];  // (2048,32,512) == (65536,512)
    const float* W      = (const float*)d_in[2];  // (512,1024)
    const float* bias   = (const float*)d_in[3];  // (512)
    const float* v      = (const float*)d_in[4];  // (512)
    float* out = (float*)d_out;                   // (32,1,2048)

    float* pre    = (float*)d_ws;                 // 32*512 floats
    float* scores = pre + B_DIM * H_DIM;          // 32*2048 floats

    precompute_kernel<<<64, 256, 0, stream>>>(hidden, W, bias, pre);
    attn_energy_kernel<<<M_DIM / 64, 512, 0, stream>>>(enc, W, pre, v, scores);
    softmax_kernel<<<B_DIM, 256, 0, stream>>>(scores, out);
}